// GATLayer_16363825398385
// MI455X (gfx1250) — compile-verified
//
#include <hip/hip_runtime.h>
#include <hip/hip_bf16.h>

typedef __attribute__((ext_vector_type(2))) float v2f;
typedef __attribute__((ext_vector_type(8))) float v8f;

#define GAT_IN_DIM 128
#define GAT_OUT_DIM 64
#define LRELU(x) ((x) > 0.0f ? (x) : 0.01f * (x))

// ---------------------------------------------------------------------------
// Kernel 1: init out/denom to 0, emax to -inf
// ---------------------------------------------------------------------------
__global__ void gat_init(float* __restrict__ out, float* __restrict__ emax,
                         float* __restrict__ denom, int n_nodes) {
    int i = blockIdx.x * blockDim.x + threadIdx.x;
    if (i < n_nodes * GAT_OUT_DIM) out[i] = 0.0f;
    if (i < n_nodes) {
        emax[i] = -__builtin_huge_valf();
        denom[i] = 0.0f;
    }
}

// ---------------------------------------------------------------------------
// Kernel 2: z = leaky_relu(h @ W), s_src = z@a_src, s_dst = z@a_dst
// One wave per 16-row tile, fp32 WMMA 16x16x4, W staged in LDS.
//
// A layout (16x4 f32): lanes 0-15 -> M=lane, v0=K0,v1=K1; lanes 16-31 -> K2,K3
// B layout (4x16 f32): lanes 0-15 -> N=lane, v0=K0,v1=K1; lanes 16-31 -> K2,K3
// C/D layout (16x16 f32): vgpr i, lanes 0-15 -> M=i, lanes 16-31 -> M=8+i
//
// LDS holds W in K-pair-interleaved form: (k,n) at Wl[(k>>1)*128 + n*2 + (k&1)]
// so each lane's B operand {W[k][n], W[k+1][n]} is ONE aligned ds_load_b64
// directly into an even VGPR pair (no repack movs before v_wmma).
// ---------------------------------------------------------------------------
__global__ __launch_bounds__(256) void gat_zgemm(
    const float* __restrict__ h, const float* __restrict__ W,
    const float* __restrict__ attn, float* __restrict__ z,
    float* __restrict__ ssrc, float* __restrict__ sdst, int n_tiles) {
    __shared__ float Wl[GAT_IN_DIM * GAT_OUT_DIM];  // 32 KB, interleaved layout
    {
        // 64 K-pairs x 64 cols; each thread stages 16 (pair,col) entries.
#pragma unroll
        for (int i = 0; i < (GAT_IN_DIM / 2 * GAT_OUT_DIM) / 256; ++i) {
            int idx = threadIdx.x + i * 256;  // idx = p*64 + n
            int p = idx >> 6;                 // K-pair index (k = 2p, 2p+1)
            int n = idx & 63;
            float2 v;
            v.x = W[(2 * p) * GAT_OUT_DIM + n];
            v.y = W[(2 * p + 1) * GAT_OUT_DIM + n];
            *(float2*)(Wl + idx * 2) = v;
        }
    }
    __syncthreads();

    const int lane = threadIdx.x & 31;
    const int wave = threadIdx.x >> 5;
    const int tile = blockIdx.x * 8 + wave;
    if (tile >= n_tiles) return;  // wave-uniform: EXEC stays all-1s for WMMA

    const int half = lane >> 4;  // 0: lanes 0-15, 1: lanes 16-31
    const int l15 = lane & 15;
    const int r0 = tile << 4;

    v8f acc[4] = {};  // 4 N-tiles of 16 cols => 64 outputs per row

    const float* hrow = h + (size_t)(r0 + l15) * GAT_IN_DIM + 2 * half;
    // base LDS offset for this lane: pair index advances by k0/2 each step
    const float* wbase = Wl + half * (GAT_OUT_DIM * 2) + l15 * 2;
#pragma unroll 4
    for (int k0 = 0; k0 < GAT_IN_DIM; k0 += 4) {
        v2f a = *(const v2f*)(hrow + k0);  // {K0,K1} or {K2,K3} for row l15
        const float* wb = wbase + (k0 >> 1) * (GAT_OUT_DIM * 2);
#pragma unroll
        for (int nt = 0; nt < 4; ++nt) {
            v2f b = *(const v2f*)(wb + nt * 32);  // one ds_load_b64
            acc[nt] = __builtin_amdgcn_wmma_f32_16x16x4_f32(
                false, a, false, b, (short)0, acc[nt], false, false);
        }
    }

    float asrc[4], adst[4];
#pragma unroll
    for (int nt = 0; nt < 4; ++nt) {
        asrc[nt] = attn[nt * 16 + l15];
        adst[nt] = attn[GAT_OUT_DIM + nt * 16 + l15];
    }

#pragma unroll
    for (int i = 0; i < 8; ++i) {
        const int row = r0 + i + half * 8;
        float ps = 0.0f, pd = 0.0f;
#pragma unroll
        for (int nt = 0; nt < 4; ++nt) {
            float v = acc[nt][i];
            v = LRELU(v);
            z[(size_t)row * GAT_OUT_DIM + nt * 16 + l15] = v;
            ps += v * asrc[nt];
            pd += v * adst[nt];
        }
        // reduce across the 16 lanes of each half-wave (xor < 16 stays in half)
#pragma unroll
        for (int m = 1; m < 16; m <<= 1) {
            ps += __shfl_xor(ps, m, 32);
            pd += __shfl_xor(pd, m, 32);
        }
        if (l15 == 0) {
            ssrc[row] = ps;
            sdst[row] = pd;
        }
    }
}

// Scalar tail for n_nodes not a multiple of 16 (not hit for 100000 nodes).
__global__ void gat_ztail(const float* __restrict__ h, const float* __restrict__ W,
                          const float* __restrict__ attn, float* __restrict__ z,
                          float* __restrict__ ssrc, float* __restrict__ sdst,
                          int row0, int n_nodes) {
    int row = row0 + blockIdx.x * blockDim.x + threadIdx.x;
    if (row >= n_nodes) return;
    float ps = 0.0f, pd = 0.0f;
    for (int c = 0; c < GAT_OUT_DIM; ++c) {
        float a = 0.0f;
        for (int k = 0; k < GAT_IN_DIM; ++k)
            a += h[(size_t)row * GAT_IN_DIM + k] * W[k * GAT_OUT_DIM + c];
        a = LRELU(a);
        z[(size_t)row * GAT_OUT_DIM + c] = a;
        ps += a * attn[c];
        pd += a * attn[GAT_OUT_DIM + c];
    }
    ssrc[row] = ps;
    sdst[row] = pd;
}

// ---------------------------------------------------------------------------
// Kernel 3: per-edge logits + segment max via monotone int-punned atomics
// ---------------------------------------------------------------------------
__device__ __forceinline__ void atomicMaxF(float* addr, float val) {
    if (val >= 0.0f)
        atomicMax((int*)addr, __float_as_int(val));
    else
        atomicMin((unsigned int*)addr, __float_as_uint(val));
}

__global__ void gat_edge_logits(const int* __restrict__ src, const int* __restrict__ dst,
                                const float* __restrict__ perf,
                                const float* __restrict__ ssrc, const float* __restrict__ sdst,
                                const float* __restrict__ w_edge, const float* __restrict__ w_m,
                                float* __restrict__ ne, float* __restrict__ emax, int n_edges) {
    int e = blockIdx.x * blockDim.x + threadIdx.x;
    if (e >= n_edges) return;
    int s = src[e], d = dst[e];
    float a = ssrc[s] + sdst[d];
    float t = a * (perf[e] * w_edge[0]);
    t = LRELU(t);
    float v = t * w_m[0];
    ne[e] = v;
    atomicMaxF(&emax[d], v);
}

// ---------------------------------------------------------------------------
// Kernel 4: emax = finite(emax) ? emax : 0   (empty segments gave -inf)
// ---------------------------------------------------------------------------
__global__ void gat_fix_emax(float* __restrict__ emax, int n_nodes) {
    int i = blockIdx.x * blockDim.x + threadIdx.x;
    if (i >= n_nodes) return;
    float m = emax[i];
    if (!__builtin_isfinite(m)) m = 0.0f;
    emax[i] = m;
}

// ---------------------------------------------------------------------------
// Kernel 5: ex = exp(ne - emax[dst]) (in place), denom[dst] += ex
// ---------------------------------------------------------------------------
__global__ void gat_edge_exp(const int* __restrict__ dst, const float* __restrict__ emax,
                             float* __restrict__ ne, float* __restrict__ denom, int n_edges) {
    int e = blockIdx.x * blockDim.x + threadIdx.x;
    if (e >= n_edges) return;
    int d = dst[e];
    float ex = expf(ne[e] - emax[d]);
    ne[e] = ex;
    atomicAdd(&denom[d], ex);
}

// ---------------------------------------------------------------------------
// Kernel 6: out[dst] += (ex/denom[dst]) * z[src] — 16 threads/edge, float4
// ---------------------------------------------------------------------------
__global__ void gat_scatter(const int* __restrict__ src, const int* __restrict__ dst,
                            const float* __restrict__ ex, const float* __restrict__ denom,
                            const float* __restrict__ z, float* __restrict__ out,
                            int n_edges) {
    unsigned tid = blockIdx.x * blockDim.x + threadIdx.x;
    int e = tid >> 4;
    if (e >= n_edges) return;
    int c4 = (tid & 15) << 2;
    int d = dst[e], s = src[e];
    float alpha = ex[e] / fmaxf(denom[d], 1e-38f);
    float4 zv = *(const float4*)(z + (size_t)s * GAT_OUT_DIM + c4);
    float* o = out + (size_t)d * GAT_OUT_DIM + c4;
    atomicAdd(o + 0, alpha * zv.x);
    atomicAdd(o + 1, alpha * zv.y);
    atomicAdd(o + 2, alpha * zv.z);
    atomicAdd(o + 3, alpha * zv.w);
}

// ---------------------------------------------------------------------------
extern "C" void kernel_launch(void* const* d_in, const int* in_sizes, int n_in,
                              void* d_out, int out_size, void* d_ws, size_t ws_size,
                              hipStream_t stream) {
    const float* h      = (const float*)d_in[0];
    const float* perf   = (const float*)d_in[1];
    const int*   src    = (const int*)d_in[2];
    const int*   dst    = (const int*)d_in[3];
    const float* W      = (const float*)d_in[4];
    const float* attn   = (const float*)d_in[5];
    const float* w_edge = (const float*)d_in[6];
    const float* w_m    = (const float*)d_in[7];
    float* out = (float*)d_out;

    const int n_nodes = in_sizes[0] / GAT_IN_DIM;
    const int n_edges = in_sizes[2];

    float* ws    = (float*)d_ws;
    float* z     = ws;                                   // n_nodes * 64
    float* ssrc  = z + (size_t)n_nodes * GAT_OUT_DIM;    // n_nodes
    float* sdst  = ssrc + n_nodes;                       // n_nodes
    float* emax  = sdst + n_nodes;                       // n_nodes
    float* denom = emax + n_nodes;                       // n_nodes
    float* ne    = denom + n_nodes;                      // n_edges (reused as ex)

    const int n_tiles = n_nodes / 16;
    const int rem = n_nodes - n_tiles * 16;

    gat_init<<<(n_nodes * GAT_OUT_DIM + 255) / 256, 256, 0, stream>>>(out, emax, denom, n_nodes);

    if (n_tiles > 0)
        gat_zgemm<<<(n_tiles + 7) / 8, 256, 0, stream>>>(h, W, attn, z, ssrc, sdst, n_tiles);
    if (rem > 0)
        gat_ztail<<<(rem + 63) / 64, 64, 0, stream>>>(h, W, attn, z, ssrc, sdst,
                                                      n_tiles * 16, n_nodes);

    gat_edge_logits<<<(n_edges + 255) / 256, 256, 0, stream>>>(
        src, dst, perf, ssrc, sdst, w_edge, w_m, ne, emax, n_edges);

    gat_fix_emax<<<(n_nodes + 255) / 256, 256, 0, stream>>>(emax, n_nodes);

    gat_edge_exp<<<(n_edges + 255) / 256, 256, 0, stream>>>(dst, emax, ne, denom, n_edges);

    long total = (long)n_edges * 16;
    gat_scatter<<<(unsigned)((total + 255) / 256), 256, 0, stream>>>(
        src, dst, ne, denom, z, out, n_edges);
}